// LinearAdderV1_30133490549219
// MI455X (gfx1250) — compile-verified
//
#include <hip/hip_runtime.h>

typedef __attribute__((ext_vector_type(2))) float v2f;
typedef __attribute__((ext_vector_type(8))) float v8f;

#define NDIM 4096

// ---------------------------------------------------------------------------
// Stage 1: row sums of input (-> sums[0..4095]) and weight (-> sums[4096..]).
// One wave per row, float4 (b128) coalesced streaming loads, wave32 butterfly
// reduction with __shfl_xor.  Grid: (512, 2) x 256 threads.
// ---------------------------------------------------------------------------
__global__ __launch_bounds__(256) void rowsum_kernel(const float* __restrict__ in0,
                                                     const float* __restrict__ in1,
                                                     float* __restrict__ sums) {
    const int wave = threadIdx.x >> 5;
    const int lane = threadIdx.x & 31;
    const int row  = blockIdx.x * 8 + wave;
    const float* __restrict__ src = blockIdx.y ? in1 : in0;
    const float4* __restrict__ rp = (const float4*)(src + (size_t)row * NDIM);

    float s = 0.0f;
#pragma unroll 4
    for (int i = lane; i < NDIM / 4; i += 32) {
        float4 v = rp[i];
        s += (v.x + v.y) + (v.z + v.w);
    }
#pragma unroll
    for (int off = 16; off > 0; off >>= 1)
        s += __shfl_xor(s, off, 32);

    if (lane == 0)
        sums[blockIdx.y * NDIM + row] = s;
}

// ---------------------------------------------------------------------------
// Stage 2: out[b,o] = (xsum[b] + wsum[o]) * bias[o], one 16x16 tile per WMMA.
//   A(16x4): [xsum[b], 1, 0, 0]   B(4x16): rows [1; wsum[o]; 0; 0]
//   D = xsum[b]*1 + 1*wsum[o]  (exact f32 add), then scale by bias[o].
// A 16x4 f32 layout: lanes 0-15 = M, VGPR0=K0, VGPR1=K1; lanes 16-31 = K2/K3.
// B 4x16  f32 layout: lanes 0-15 = rows K0/K1, lanes 16-31 = rows K2/K3.
// D 16x16 f32 layout: VGPR r -> row M = r + 8*(lane>=16), lane%16 = column.
// Each wave: one row-strip of 16 b-rows, 8 column tiles (8 KB of output).
// Grid: 1024 blocks x 256 threads (8 waves) -> 8192 waves, 65536 tiles.
// ---------------------------------------------------------------------------
__global__ __launch_bounds__(256) void linadder_tile_kernel(
        const float* __restrict__ xsum,
        const float* __restrict__ wsum,
        const float* __restrict__ bias,
        float* __restrict__ out) {
    const int wave = threadIdx.x >> 5;
    const int lane = threadIdx.x & 31;
    const int l16  = lane & 15;
    const int half = lane >> 4;          // 0: lanes 0-15, 1: lanes 16-31

    const int gw       = blockIdx.x * 8 + wave;   // 0..8191
    const int rowStrip = gw >> 5;                 // 0..255  (b tile)
    const int colGroup = gw & 31;                 // 0..31   (8 o tiles each)
    const int b_base   = rowStrip * 16;

    // A operand: K=0 -> xsum[b], K=1 -> 1.0; K=2,3 (upper half-wave) -> 0.
    const float xv = xsum[b_base + l16];          // uniform-ish, L2-hot
    v2f a;
    a.x = half ? 0.0f : xv;
    a.y = half ? 0.0f : 1.0f;

    const float bk0 = half ? 0.0f : 1.0f;         // B row K=0 = ones

#pragma unroll
    for (int t = 0; t < 8; ++t) {
        const int o_base = (colGroup * 8 + t) * 16;

        const float wv = wsum[o_base + l16];
        v2f b;
        b.x = bk0;                                // K=0 row: 1.0
        b.y = half ? 0.0f : wv;                   // K=1 row: wsum[o]

        v8f c = {};
        // D = A x B + 0  ->  D[m,n] = xsum[b_base+m] + wsum[o_base+n]
        v8f d = __builtin_amdgcn_wmma_f32_16x16x4_f32(
            /*neg_a=*/false, a, /*neg_b=*/false, b,
            /*c_mod=*/(short)0, c, /*reuse_a=*/false, /*reuse_b=*/false);

        const float bs = bias[o_base + l16];      // lane's column o
        float* __restrict__ p =
            out + (size_t)(b_base + half * 8) * NDIM + o_base + l16;

#pragma unroll
        for (int r = 0; r < 8; ++r)
            p[(size_t)r * NDIM] = d[r] * bs;      // 2x64B contiguous per store
    }
}

extern "C" void kernel_launch(void* const* d_in, const int* in_sizes, int n_in,
                              void* d_out, int out_size, void* d_ws, size_t ws_size,
                              hipStream_t stream) {
    const float* input  = (const float*)d_in[0];   // (B, IN)  f32
    const float* weight = (const float*)d_in[1];   // (OUT, IN) f32
    const float* bias   = (const float*)d_in[2];   // (OUT,)   f32
    float*       out    = (float*)d_out;           // (B, OUT) f32
    float*       sums   = (float*)d_ws;            // [0..4095]=xsum, [4096..8191]=wsum

    (void)in_sizes; (void)n_in; (void)out_size; (void)ws_size;

    rowsum_kernel<<<dim3(NDIM / 8, 2), 256, 0, stream>>>(input, weight, sums);
    linadder_tile_kernel<<<dim3(1024), 256, 0, stream>>>(sums, sums + NDIM, bias, out);
}